// FPS_KNN_PCT_24300924961394
// MI455X (gfx1250) — compile-verified
//
#include <hip/hip_runtime.h>
#include <hip/hip_bf16.h>

// ---------------------------------------------------------------------------
// PCT sample-and-group + 2x (1x1 conv -> BN(train) -> ReLU) -> max over K
// MI455X (gfx1250, wave32).
//   * GEMMs on V_WMMA_F32_16X16X4_F32 (reference fp32 precision).
//   * B operand streamed HBM -> LDS via global_load_async_to_lds_b128
//     (double-buffered, s_wait_asynccnt), read back with ds loads.
//   * Block owns all M=256 rows (weights are L2-resident), so the 128 MB
//     activation matrix is read from HBM exactly once per GEMM.
// ---------------------------------------------------------------------------

#define BATCH    8
#define NPTS     4096
#define S_NPOINT 512
#define KNN      32
#define DIN      64
#define D2       128            // 2*DIN
#define OUTC     256
#define NT       (BATCH * S_NPOINT * KNN)   // 131072 columns of the big GEMMs
#define NGROUP   (BATCH * S_NPOINT)         // 4096
#define BN_EPS   1e-5f

typedef float v2f __attribute__((ext_vector_type(2)));
typedef float v8f __attribute__((ext_vector_type(8)));

__device__ __forceinline__ void wait_asynccnt_1() {
#if __has_builtin(__builtin_amdgcn_s_wait_asynccnt)
    __builtin_amdgcn_s_wait_asynccnt(1);
#else
    asm volatile("s_wait_asynccnt 0x1" ::: "memory");
#endif
}
__device__ __forceinline__ void wait_asynccnt_0() {
#if __has_builtin(__builtin_amdgcn_s_wait_asynccnt)
    __builtin_amdgcn_s_wait_asynccnt(0);
#else
    asm volatile("s_wait_asynccnt 0x0" ::: "memory");
#endif
}

// ---------------------------------------------------------------------------
// 1) Farthest point sampling: one block per batch, 512 serial argmax rounds.
// ---------------------------------------------------------------------------
__global__ __launch_bounds__(256) void fps_kernel(const float* __restrict__ xyz,
                                                  int* __restrict__ fps_idx,
                                                  float* __restrict__ new_xyz) {
    const int b = blockIdx.x;
    const int t = threadIdx.x;
    const float* X = xyz + (size_t)b * NPTS * 3;

    float px[16], py[16], pz[16], dist[16];
#pragma unroll
    for (int j = 0; j < 16; ++j) {
        int p = t + 256 * j;
        px[j] = X[p * 3 + 0];
        py[j] = X[p * 3 + 1];
        pz[j] = X[p * 3 + 2];
        dist[j] = 1e10f;
    }

    __shared__ float sval[256];
    __shared__ int   sidx[256];
    __shared__ float sc[3];

    int far = 0;
    for (int s = 0; s < S_NPOINT; ++s) {
        if (t == 0) {
            fps_idx[b * S_NPOINT + s] = far;
            float cx = X[far * 3 + 0], cy = X[far * 3 + 1], cz = X[far * 3 + 2];
            sc[0] = cx; sc[1] = cy; sc[2] = cz;
            float* o = new_xyz + (size_t)(b * S_NPOINT + s) * 3;
            o[0] = cx; o[1] = cy; o[2] = cz;
        }
        __syncthreads();
        const float cx = sc[0], cy = sc[1], cz = sc[2];

        float best = -1.0f; int bidx = 0;
#pragma unroll
        for (int j = 0; j < 16; ++j) {
            float dx = px[j] - cx, dy = py[j] - cy, dz = pz[j] - cz;
            float d  = dx * dx + dy * dy + dz * dz;
            float nd = fminf(dist[j], d);
            dist[j]  = nd;
            if (nd > best) { best = nd; bidx = t + 256 * j; }  // strict > : first max
        }
        sval[t] = best; sidx[t] = bidx;
        __syncthreads();
        for (int off = 128; off > 0; off >>= 1) {
            if (t < off) {
                float ov = sval[t + off]; int oi = sidx[t + off];
                if (ov > sval[t] || (ov == sval[t] && oi < sidx[t])) {
                    sval[t] = ov; sidx[t] = oi;
                }
            }
            __syncthreads();
        }
        far = sidx[0];
    }
}

// ---------------------------------------------------------------------------
// 2) Gather feature vectors of the sampled points: new_pts[g][c]
// ---------------------------------------------------------------------------
__global__ __launch_bounds__(256) void gather_pts_kernel(const float* __restrict__ points,
                                                         const int* __restrict__ fps_idx,
                                                         float* __restrict__ new_pts) {
    int u = blockIdx.x * 256 + threadIdx.x;
    if (u >= NGROUP * DIN) return;
    int c = u % DIN;
    int g = u / DIN;
    int b = g / S_NPOINT;
    int p = fps_idx[g];
    new_pts[u] = points[((size_t)b * NPTS + p) * DIN + c];
}

// ---------------------------------------------------------------------------
// 3) 32-NN per query: one wave per query, distances staged in LDS,
//    32 rounds of wave-butterfly argmin (ties -> smaller index).
// ---------------------------------------------------------------------------
__global__ __launch_bounds__(64) void knn_kernel(const float* __restrict__ xyz,
                                                 const float* __restrict__ new_xyz,
                                                 int* __restrict__ knn_idx) {
    __shared__ float sd[2][NPTS];   // 32 KB
    const int w    = threadIdx.x >> 5;
    const int lane = threadIdx.x & 31;
    const int q    = blockIdx.x * 2 + w;
    const int b    = q >> 9;
    const float* X = xyz + (size_t)b * NPTS * 3;

    const float qx = new_xyz[q * 3 + 0];
    const float qy = new_xyz[q * 3 + 1];
    const float qz = new_xyz[q * 3 + 2];

    float* D = sd[w];
#pragma unroll 4
    for (int j = 0; j < NPTS / 32; ++j) {
        int p = lane + 32 * j;
        float dx = X[p * 3 + 0] - qx;
        float dy = X[p * 3 + 1] - qy;
        float dz = X[p * 3 + 2] - qz;
        D[p] = dx * dx + dy * dy + dz * dz;
    }
    __syncthreads();

    for (int r = 0; r < KNN; ++r) {
        float best = 3.4e38f; int bi = lane;
        for (int j = 0; j < NPTS / 32; ++j) {
            int p = lane + 32 * j;
            float v = D[p];
            if (v < best) { best = v; bi = p; }
        }
#pragma unroll
        for (int m = 16; m > 0; m >>= 1) {
            float ov = __shfl_xor(best, m, 32);
            int   oi = __shfl_xor(bi,   m, 32);
            if (ov < best || (ov == best && oi < bi)) { best = ov; bi = oi; }
        }
        if (lane == 0) {
            knn_idx[q * KNN + r] = bi;
            D[bi] = 3.4e38f;
        }
        __syncthreads();
    }
}

// ---------------------------------------------------------------------------
// 4) Build feature matrix transposed for the GEMM: featsT[row=0..127][col].
// ---------------------------------------------------------------------------
__global__ __launch_bounds__(256) void feats_kernel(const float* __restrict__ points,
                                                    const float* __restrict__ new_pts,
                                                    const int* __restrict__ knn_idx,
                                                    float* __restrict__ featsT) {
    const int g  = blockIdx.x;            // 0..4095
    const int k  = threadIdx.x & 31;
    const int c0 = threadIdx.x >> 5;      // 0..7
    const int b  = g >> 9;
    const int p  = knn_idx[g * KNN + k];
    const float* P  = points  + ((size_t)b * NPTS + p) * DIN;
    const float* NP = new_pts + (size_t)g * DIN;
    const size_t col = (size_t)g * KNN + k;
    for (int c = c0; c < DIN; c += 8) {
        float np = NP[c];
        featsT[(size_t)c * NT + col]         = P[c] - np;
        featsT[(size_t)(DIN + c) * NT + col] = np;
    }
}

// ---------------------------------------------------------------------------
// 5) WMMA fp32 GEMM:  C[256 x NT] = A[256 x K] * B[K x NT], row-major.
//    Block (8 waves) covers all 256 M rows x 32 N cols.  B strip is staged
//    HBM -> LDS with global_load_async_to_lds_b128 (one 16B segment per
//    thread => one async instruction per wave per stage), double-buffered
//    with s_wait_asynccnt.  Wave w owns m-tiles {2w, 2w+1} x n-tiles {0,1}:
//    4 accumulators, 4 v_wmma_f32_16x16x4_f32 per K-step.
//    WMMA fragment layouts (lo=lane&15, hi=lane>>4):
//      A 16x4 : M = base+lo, K pair {k+2hi, k+2hi+1} -> one float2 load
//      B 4x16 : same K pair, N = base+lo
//      C 16x16: vgpr r -> row base+r+8hi, col base+lo
//    BN template: fuse y = relu(x*a+d) (coeffs in LDS) onto B loads (GEMM2).
// ---------------------------------------------------------------------------
template <bool BN>
__global__ __launch_bounds__(256) void wmma_gemm_kernel(const float* __restrict__ A,
                                                        const float* __restrict__ Bm,
                                                        float* __restrict__ C,
                                                        int K,
                                                        const float* __restrict__ bnA,
                                                        const float* __restrict__ bnD) {
    __shared__ float bbuf[2][32 * 32];   // double-buffered B chunk [32 K][32 N]
    __shared__ float sA[OUTC], sD[OUTC]; // BN affine per B-row (BN path only)

    const int t     = threadIdx.x;
    const int lane  = t & 31;
    const int w     = t >> 5;
    const int hi    = lane >> 4;
    const int lo    = lane & 15;
    const int nBase = blockIdx.x * 32;

    if (BN) {
        if (t < K) { sA[t] = bnA[t]; sD[t] = bnD[t]; }
    }

    // Async stage: thread t moves 16B; row = t>>3 (0..31), seg = (t&7)*4.
    const int   srow = t >> 3;
    const int   scol = (t & 7) * 4;
    const float* gsrc0 = Bm + (size_t)srow * NT + nBase + scol;

    auto stage = [&](int bufsel, int k0) {
        const float* gp = gsrc0 + (size_t)k0 * NT;
        unsigned int lp = (unsigned int)(size_t)&bbuf[bufsel][srow * 32 + scol];
        unsigned long long ga = (unsigned long long)(size_t)gp;
        asm volatile("global_load_async_to_lds_b128 %0, %1, off"
                     :: "v"(lp), "v"(ga) : "memory");
    };

    v8f z = {0.f,0.f,0.f,0.f,0.f,0.f,0.f,0.f};
    v8f acc00 = z, acc01 = z, acc10 = z, acc11 = z;

    const float* A0 = A + (size_t)(32 * w + lo) * K + 2 * hi;       // m-tile 2w
    const float* A1 = A0 + (size_t)16 * K;                          // m-tile 2w+1

    stage(0, 0);
    const int NC = K / 32;
    for (int c = 0; c < NC; ++c) {
        const int buf = c & 1;
        const int k0  = c * 32;
        if (c + 1 < NC) {
            stage(buf ^ 1, k0 + 32);
            wait_asynccnt_1();          // chunk c landed (in-order per wave)
        } else {
            wait_asynccnt_0();
        }
        __syncthreads();                // all waves' segments visible

        const float* Bl = &bbuf[buf][0];
#pragma unroll
        for (int ks = 0; ks < 8; ++ks) {
            const int kk = ks * 4 + 2 * hi;            // local K pair {kk, kk+1}
            v2f a0 = *(const v2f*)(A0 + k0 + ks * 4);
            v2f a1 = *(const v2f*)(A1 + k0 + ks * 4);
            float b0a = Bl[kk * 32 + lo];
            float b0b = Bl[(kk + 1) * 32 + lo];
            float b1a = Bl[kk * 32 + 16 + lo];
            float b1b = Bl[(kk + 1) * 32 + 16 + lo];
            if (BN) {
                const int r = k0 + kk;
                float sa0 = sA[r], sd0 = sD[r];
                float sa1 = sA[r + 1], sd1 = sD[r + 1];
                b0a = fmaxf(0.f, b0a * sa0 + sd0);
                b0b = fmaxf(0.f, b0b * sa1 + sd1);
                b1a = fmaxf(0.f, b1a * sa0 + sd0);
                b1b = fmaxf(0.f, b1b * sa1 + sd1);
            }
            v2f b0 = {b0a, b0b};
            v2f b1 = {b1a, b1b};
            acc00 = __builtin_amdgcn_wmma_f32_16x16x4_f32(false, a0, false, b0, (short)0, acc00, false, false);
            acc01 = __builtin_amdgcn_wmma_f32_16x16x4_f32(false, a0, false, b1, (short)0, acc01, false, false);
            acc10 = __builtin_amdgcn_wmma_f32_16x16x4_f32(false, a1, false, b0, (short)0, acc10, false, false);
            acc11 = __builtin_amdgcn_wmma_f32_16x16x4_f32(false, a1, false, b1, (short)0, acc11, false, false);
        }
        __syncthreads();                // done reading buf before it is re-staged
    }

    const int n0  = nBase + lo;
    const int n1  = nBase + 16 + lo;
    const int m0r = 32 * w + 8 * hi;        // + r  (m-tile 2w)
    const int m1r = m0r + 16;               // + r  (m-tile 2w+1)
#pragma unroll
    for (int r = 0; r < 8; ++r) {
        C[(size_t)(m0r + r) * NT + n0] = acc00[r];
        C[(size_t)(m0r + r) * NT + n1] = acc01[r];
        C[(size_t)(m1r + r) * NT + n0] = acc10[r];
        C[(size_t)(m1r + r) * NT + n1] = acc11[r];
    }
}

// ---------------------------------------------------------------------------
// 6) BatchNorm (training) stats per channel over NT samples -> affine a,d.
// ---------------------------------------------------------------------------
__global__ __launch_bounds__(256) void bnstats_kernel(const float* __restrict__ X,
                                                      const float* __restrict__ gamma,
                                                      const float* __restrict__ beta,
                                                      float* __restrict__ cA,
                                                      float* __restrict__ cD) {
    const int row = blockIdx.x;
    const int t   = threadIdx.x;
    const float* R = X + (size_t)row * NT;
    float s = 0.f, sq = 0.f;
    for (int j = t; j < NT; j += 256) { float v = R[j]; s += v; sq += v * v; }
    __shared__ float ss[256], s2[256];
    ss[t] = s; s2[t] = sq;
    __syncthreads();
    for (int off = 128; off > 0; off >>= 1) {
        if (t < off) { ss[t] += ss[t + off]; s2[t] += s2[t + off]; }
        __syncthreads();
    }
    if (t == 0) {
        float mean = ss[0] / (float)NT;
        float var  = s2[0] / (float)NT - mean * mean;   // biased, like jnp.var
        float a    = gamma[row] * rsqrtf(var + BN_EPS);
        cA[row] = a;
        cD[row] = beta[row] - mean * a;
    }
}

// ---------------------------------------------------------------------------
// 7) Epilogue: relu(BN2(X2)) then max over K=32, write x [B,S,OUT].
// ---------------------------------------------------------------------------
__global__ __launch_bounds__(256) void finalize_kernel(const float* __restrict__ X2,
                                                       const float* __restrict__ cA,
                                                       const float* __restrict__ cD,
                                                       float* __restrict__ out) {
    const int g = blockIdx.x;    // group 0..4095
    const int c = threadIdx.x;   // channel 0..255
    const float a = cA[c], d = cD[c];
    const float* R = X2 + (size_t)c * NT + (size_t)g * KNN;
    float m = 0.f;               // relu output >= 0
#pragma unroll
    for (int k = 0; k < KNN; ++k) {
        m = fmaxf(m, fmaxf(0.f, R[k] * a + d));
    }
    out[(size_t)g * OUTC + c] = m;
}

// ---------------------------------------------------------------------------
extern "C" void kernel_launch(void* const* d_in, const int* in_sizes, int n_in,
                              void* d_out, int out_size, void* d_ws, size_t ws_size,
                              hipStream_t stream) {
    const float* xyz    = (const float*)d_in[0];
    const float* points = (const float*)d_in[1];
    const float* w1     = (const float*)d_in[2];
    const float* g1     = (const float*)d_in[3];
    const float* b1     = (const float*)d_in[4];
    const float* w2     = (const float*)d_in[5];
    const float* g2     = (const float*)d_in[6];
    const float* b2     = (const float*)d_in[7];

    float* out     = (float*)d_out;
    float* new_xyz = out;                         // [B,S,3]  = 12288 floats
    float* x_out   = out + (size_t)NGROUP * 3;    // [B,S,OUT]

    // workspace carve-up (floats)
    float* wsf    = (float*)d_ws;
    size_t o      = 0;
    float* featsT = wsf + o; o += (size_t)D2  * NT;   // 128 x 131072
    float* X1     = wsf + o; o += (size_t)OUTC * NT;  // 256 x 131072 (raw conv1)
    float* X2     = wsf + o; o += (size_t)OUTC * NT;  // 256 x 131072 (raw conv2)
    float* newp   = wsf + o; o += (size_t)NGROUP * DIN;
    float* cA1    = wsf + o; o += OUTC;
    float* cD1    = wsf + o; o += OUTC;
    float* cA2    = wsf + o; o += OUTC;
    float* cD2    = wsf + o; o += OUTC;
    int*   fpsIdx = (int*)(wsf + o); o += NGROUP;
    int*   knnIdx = (int*)(wsf + o); o += (size_t)NGROUP * KNN;

    // 1) FPS (also emits new_xyz directly into d_out)
    fps_kernel<<<BATCH, 256, 0, stream>>>(xyz, fpsIdx, new_xyz);
    // 2) gather sampled point features
    gather_pts_kernel<<<(NGROUP * DIN + 255) / 256, 256, 0, stream>>>(points, fpsIdx, newp);
    // 3) 32-NN
    knn_kernel<<<NGROUP / 2, 64, 0, stream>>>(xyz, new_xyz, knnIdx);
    // 4) feature matrix (transposed for GEMM)
    feats_kernel<<<NGROUP, 256, 0, stream>>>(points, newp, knnIdx, featsT);

    // 5) conv1 as WMMA GEMM: [256 x 128] x [128 x 131072]
    wmma_gemm_kernel<false><<<dim3(NT / 32), 256, 0, stream>>>(w1, featsT, X1, D2, nullptr, nullptr);
    // 6) BN1 stats -> affine
    bnstats_kernel<<<OUTC, 256, 0, stream>>>(X1, g1, b1, cA1, cD1);
    // 7) conv2 as WMMA GEMM with fused relu(BN1) on the B operand
    wmma_gemm_kernel<true><<<dim3(NT / 32), 256, 0, stream>>>(w2, X1, X2, OUTC, cA1, cD1);
    // 8) BN2 stats
    bnstats_kernel<<<OUTC, 256, 0, stream>>>(X2, g2, b2, cA2, cD2);
    // 9) relu(BN2) + max over K -> output
    finalize_kernel<<<NGROUP, 256, 0, stream>>>(X2, cA2, cD2, x_out);
}